// GVI_multi_classification_40312563040561
// MI455X (gfx1250) — compile-verified
//
#include <hip/hip_runtime.h>
#include <stdint.h>
#include <stddef.h>

// ---------------------------------------------------------------------------
// Problem constants (match reference)
// ---------------------------------------------------------------------------
#define B_DIM 4096
#define C_DIM 1000
#define NQ    50
#define QP    64          // Q padded to a multiple of 16 for WMMA tiles

// ws layout (floats):
//   [0..63]     GH roots (50 valid, rest 0)
//   [64..127]   GH weights
//   [128..191]  CONST_REMOVE[q] = log(0.5*erfc(-x_q) + 0.001)
//   [192]       reg accumulator  (sum mean_pred^2)
//   [193]       sum-L accumulator
//   [256 ..)    trq[B][QP]  (WMMA output), B*QP floats
#define WS_ROOTS   0
#define WS_WGT     64
#define WS_CR      128
#define WS_REG     192
#define WS_SUML    193
#define WS_TRQ     256

typedef float    v2f   __attribute__((ext_vector_type(2)));
typedef float    v8f   __attribute__((ext_vector_type(8)));
typedef uint32_t u32x4 __attribute__((ext_vector_type(4)));
typedef int      i32x4 __attribute__((ext_vector_type(4)));
typedef int      i32x8 __attribute__((ext_vector_type(8)));

#if defined(__has_builtin)
# if __has_builtin(__builtin_amdgcn_tensor_load_to_lds)
#  define HAVE_TDM 1
# endif
#endif
#ifndef HAVE_TDM
# define HAVE_TDM 0
#endif

// ---------------------------------------------------------------------------
// Kernel 1: deterministic device-side Gauss-Hermite (physicists') nodes,
// weights and CONST_REMOVE, via orthonormal-recurrence Newton (NR gauher).
// Also zeroes the two global accumulators every call.
// ---------------------------------------------------------------------------
__global__ void gh_init_kernel(float* ws) {
  if (threadIdx.x != 0 || blockIdx.x != 0) return;
  ws[WS_REG]  = 0.0f;
  ws[WS_SUML] = 0.0f;

  const int n = NQ;
  double xr[32];
  double z = 0.0, pp = 0.0;
  const double pim4 = 0.7511255444649425;   // pi^(-1/4)
  for (int i = 0; i < (n + 1) / 2; ++i) {
    if      (i == 0) z = sqrt((double)(2 * n + 1)) -
                         1.85575 * pow((double)(2 * n + 1), -0.16666666666666666);
    else if (i == 1) z -= 1.14 * pow((double)n, 0.426) / z;
    else if (i == 2) z = 1.86 * z - 0.86 * xr[0];
    else if (i == 3) z = 1.91 * z - 0.91 * xr[1];
    else             z = 2.0 * z - xr[i - 2];
    for (int it = 0; it < 200; ++it) {
      double p1 = pim4, p2 = 0.0;
      for (int j = 1; j <= n; ++j) {
        double p3 = p2; p2 = p1;
        p1 = z * sqrt(2.0 / (double)j) * p2 - sqrt((double)(j - 1) / (double)j) * p3;
      }
      pp = sqrt(2.0 * (double)n) * p2;
      double z1 = z;
      z = z1 - p1 / pp;
      if (fabs(z - z1) < 3.0e-14) break;
    }
    xr[i] = z;
    double wgt = 2.0 / (pp * pp);
    ws[WS_ROOTS + i]         = (float)z;
    ws[WS_ROOTS + n - 1 - i] = (float)(-z);
    ws[WS_WGT + i]           = (float)wgt;
    ws[WS_WGT + n - 1 - i]   = (float)wgt;
  }
  for (int q = n; q < QP; ++q) {
    ws[WS_ROOTS + q] = 0.0f; ws[WS_WGT + q] = 0.0f; ws[WS_CR + q] = 0.0f;
  }
  for (int q = 0; q < n; ++q) {
    double xq = (double)ws[WS_ROOTS + q];
    // ndtr(sqrt(2)*x) = 0.5*erfc(-x)
    double nd = 0.5 * erfc(-xq);
    ws[WS_CR + q] = (float)log(nd + 0.001);
  }
}

// ---------------------------------------------------------------------------
// Kernel 2: trq_j[b,q] = sqrt2*sq_y[b]*root[q] + mu_y[b]  as a rank-2 matmul
// [16x4] x [4x16] per wave using V_WMMA_F32_16X16X4_F32.
// A[m,0]=sqrt2*sq_y, A[m,1]=mu_y, A[m,2..3]=0 ; B[0,n]=root, B[1,n]=1, rest 0.
// 32-bit A layout: VGPR0 = k0 (lanes 0-15) / k2 (16-31); VGPR1 = k1 / k3.
// Operand build is branchless: all 32 lanes load (upper lanes duplicate the
// lower-lane addresses), then v_cndmask zeroes the k=2/k=3 rows. WMMA then
// executes with EXEC all-ones as the ISA requires.
// ---------------------------------------------------------------------------
__launch_bounds__(32)
__global__ void trq_wmma_kernel(const float* __restrict__ mean_pred,
                                const float* __restrict__ sq_trq,
                                const int*   __restrict__ y,
                                float*       __restrict__ ws) {
  const float* roots = ws + WS_ROOTS;
  float*       trq   = ws + WS_TRQ;
  const int lane = threadIdx.x;
  const int bt = blockIdx.x >> 2;       // B/16 = 256 row tiles
  const int qt = blockIdx.x & 3;        // QP/16 = 4 col tiles
  const int b0 = bt * 16, q0 = qt * 16;
  const bool lo = lane < 16;
  const int  l  = lane & 15;

  // unconditional loads (upper half duplicates lower-half addresses)
  const int row = b0 + l;
  const int yv  = y[row];
  const size_t off = (size_t)row * C_DIM + (size_t)yv;
  const float sqv = sq_trq[off];
  const float muv = mean_pred[off];
  const float rtv = roots[q0 + l];

  // branchless operand build: k=2 / k=3 rows forced to zero
  const float a0  = lo ? (1.41421356237309515f * sqv) : 0.0f;  // A[m,0]
  const float a1  = lo ? muv : 0.0f;                           // A[m,1]
  const float bv0 = lo ? rtv : 0.0f;                           // B[0,n]
  const float bv1 = lo ? 1.0f : 0.0f;                          // B[1,n]

  v2f A;  A[0]  = a0;  A[1]  = a1;
  v2f Bm; Bm[0] = bv0; Bm[1] = bv1;
  v8f Cz = {0.f, 0.f, 0.f, 0.f, 0.f, 0.f, 0.f, 0.f};
  v8f D = __builtin_amdgcn_wmma_f32_16x16x4_f32(
      /*neg_a=*/false, A, /*neg_b=*/false, Bm,
      /*c_mod=*/(short)0, Cz, /*reuse_a=*/false, /*reuse_b=*/false);

  const int M0 = lo ? 0 : 8;   // D VGPR r: lanes 0-15 -> M=r, lanes 16-31 -> M=r+8
#pragma unroll
  for (int r = 0; r < 8; ++r) {
    trq[(size_t)(b0 + M0 + r) * QP + (size_t)(q0 + l)] = D[r];
  }
}

// ---------------------------------------------------------------------------
// TDM: DMA one contiguous float row (nelem elements) from global into LDS.
// ---------------------------------------------------------------------------
#if HAVE_TDM
__device__ inline void tdm_load_row(const void* gsrc, uint32_t lds_byte_off,
                                    int nelem) {
  const uint64_t ga = (uint64_t)(uintptr_t)gsrc;
  u32x4 g0;
  g0[0] = 1u;                                            // count=1, user desc
  g0[1] = lds_byte_off;                                  // lds_addr
  g0[2] = (uint32_t)(ga & 0xFFFFFFFFu);                  // global_addr[31:0]
  g0[3] = (uint32_t)((ga >> 32) & 0x01FFFFFFu) | (2u << 30);  // addr hi | type=2
  i32x8 g1;
  const uint32_t dim0 = (uint32_t)nelem;
  g1[0] = (int)(2u << 16);                               // data_size = 4 bytes
  g1[1] = (int)((dim0 & 0xFFFFu) << 16);                 // tensor_dim0[15:0]
  g1[2] = (int)(((dim0 >> 16) & 0xFFFFu) | (1u << 16));  // dim0 hi | tensor_dim1=1
  g1[3] = (int)((dim0 & 0xFFFFu) << 16);                 // tile_dim0 = nelem
  g1[4] = 1;                                             // tile_dim1=1, tile_dim2=0
  g1[5] = (int)dim0;                                     // tensor_dim0_stride lo
  g1[6] = 0;
  g1[7] = 0;
  i32x4 gz = {0, 0, 0, 0};
#if defined(__clang_major__) && (__clang_major__ >= 23)
  i32x8 gz8 = {0, 0, 0, 0, 0, 0, 0, 0};
  __builtin_amdgcn_tensor_load_to_lds(g0, g1, gz, gz, gz8, 0);
#else
  __builtin_amdgcn_tensor_load_to_lds(g0, g1, gz, gz, 0);
#endif
}
#endif

__device__ inline float wave_reduce_add(float v) {
  v += __shfl_xor(v, 16);
  v += __shfl_xor(v, 8);
  v += __shfl_xor(v, 4);
  v += __shfl_xor(v, 2);
  v += __shfl_xor(v, 1);
  return v;
}

// ndtr(x) = 0.5*erfc(-x/sqrt2) via A&S 7.1.26 (|err| ~ 1.5e-7, well under the
// +0.001 smoothing), using hardware v_exp_f32.
__device__ inline float ndtr_fast(float x) {
  const float z  = -x * 0.70710678118654752f;
  const float az = fabsf(z);
  const float t  = 1.0f / fmaf(0.3275911f, az, 1.0f);
  float poly = fmaf(t, 1.061405429f, -1.453152027f);
  poly = fmaf(t, poly, 1.421413741f);
  poly = fmaf(t, poly, -0.284496736f);
  poly = fmaf(t, poly, 0.254829592f);
  poly *= t;
  const float e       = __expf(-az * az);
  const float erfc_az = poly * e;
  const float erfc_z  = (z < 0.0f) ? (2.0f - erfc_az) : erfc_az;
  return 0.5f * erfc_z;
}

// ---------------------------------------------------------------------------
// Kernel 3: main. One block (256 thr = 8 wave32) per example b.
// Rows staged to LDS by the Tensor Data Mover (wave 0 only, scalar-guarded);
// each thread owns ~4 c values, the 50 trq values and 50 q-accumulators live
// in registers; wave32 butterfly + LDS cross-wave reduction afterwards.
// ---------------------------------------------------------------------------
__launch_bounds__(256)
__global__ void gvi_main_kernel(const float* __restrict__ mean_pred,
                                const float* __restrict__ sq_trq,
                                const int*   __restrict__ y,
                                float*       __restrict__ ws,
                                int use_ws_trq) {
  __shared__ float s_m[C_DIM];
  __shared__ float s_s[C_DIM];
  __shared__ float s_trq[QP];
  __shared__ float s_wred[8][NQ];
  __shared__ float s_reg[8];

  const int b    = blockIdx.x;
  const int tid  = threadIdx.x;
  const int wave = tid >> 5;
  const int lane = tid & 31;
  const size_t rowoff = (size_t)b * C_DIM;

#if HAVE_TDM
  // Scalar (SCC) branch: readfirstlane makes the wave index an SGPR value, so
  // waves 1..7 skip this block entirely. TDM ignores EXEC, so an EXEC-masked
  // guard would not prevent duplicate DMA issues — a scalar branch does.
  {
    const uint32_t wave_sc = (uint32_t)__builtin_amdgcn_readfirstlane(tid) >> 5;
    if (wave_sc == 0) {
      tdm_load_row(mean_pred + rowoff, (uint32_t)(uintptr_t)&s_m[0], C_DIM);
      tdm_load_row(sq_trq + rowoff,    (uint32_t)(uintptr_t)&s_s[0], C_DIM);
      __builtin_amdgcn_s_wait_tensorcnt(0);
    }
  }
#else
  for (int c = tid; c < C_DIM; c += 256) {
    s_m[c] = mean_pred[rowoff + c];
    s_s[c] = sq_trq[rowoff + c];
  }
#endif

  if (tid < QP) {
    if (use_ws_trq) {
      s_trq[tid] = ws[WS_TRQ + (size_t)b * QP + tid];
    } else {
      const int yv = y[b];
      const float mu = mean_pred[rowoff + yv];
      const float sq = sq_trq[rowoff + yv];
      s_trq[tid] = (tid < NQ)
                       ? fmaf(1.41421356237309515f * sq, ws[WS_ROOTS + tid], mu)
                       : 0.0f;
    }
  }
  __syncthreads();

  // hoist the 50 trq values into registers: hot loop is pure VALU/TRANS
  float rq[NQ];
#pragma unroll
  for (int q = 0; q < NQ; ++q) rq[q] = s_trq[q];

  float acc[NQ];
#pragma unroll
  for (int q = 0; q < NQ; ++q) acc[q] = 0.0f;
  float racc = 0.0f;

  for (int c = tid; c < C_DIM; c += 256) {
    const float m = s_m[c];
    const float s = s_s[c];
    racc = fmaf(m, m, racc);
    const float inv = 1.0f / s;
#pragma unroll
    for (int q = 0; q < NQ; ++q) {
      const float x = (rq[q] - m) * inv;
      const float p = ndtr_fast(x);
      acc[q] += __logf(p + 0.001f);
    }
  }

  // wave-level reduction (all 50 q sums + reg)
#pragma unroll
  for (int q = 0; q < NQ; ++q) acc[q] = wave_reduce_add(acc[q]);
  racc = wave_reduce_add(racc);

  if (lane == 0) {
#pragma unroll
    for (int q = 0; q < NQ; ++q) s_wred[wave][q] = acc[q];
    s_reg[wave] = racc;
  }
  __syncthreads();

  if (wave == 0) {
    float part = 0.0f;
    if (lane < NQ) {
      float fsum = 0.0f;
#pragma unroll
      for (int w2 = 0; w2 < 8; ++w2) fsum += s_wred[w2][lane];
      part += __expf(fsum - ws[WS_CR + lane]) * ws[WS_WGT + lane];
    }
    const int q2 = lane + 32;
    if (q2 < NQ) {
      float fsum = 0.0f;
#pragma unroll
      for (int w2 = 0; w2 < 8; ++w2) fsum += s_wred[w2][q2];
      part += __expf(fsum - ws[WS_CR + q2]) * ws[WS_WGT + q2];
    }
    part = wave_reduce_add(part);
    if (lane == 0) {
      const float S = part * 0.56418958354775628f;   // / sqrt(pi)
      float regv = 0.0f;
#pragma unroll
      for (int w2 = 0; w2 < 8; ++w2) regv += s_reg[w2];
      const float LOG_1_MINUS_EPS = -0.0100503358535014412f;  // log(0.99)
      const float LOG_1_OVER_EPS  = -11.5119249646366454f;    // log(0.01/999)
      const float L = -(LOG_1_MINUS_EPS * S + LOG_1_OVER_EPS * (1.0f - S));
      atomicAdd(&ws[WS_SUML], L);
      atomicAdd(&ws[WS_REG], regv);
    }
  }
}

// ---------------------------------------------------------------------------
// Kernel 4: finalize -> out[0]=sum L, out[1]=sqrt(relu(tot_trace + reg))
// ---------------------------------------------------------------------------
__global__ void finalize_kernel(const float* __restrict__ tot_trace,
                                const float* __restrict__ ws,
                                float* __restrict__ out) {
  if (threadIdx.x == 0 && blockIdx.x == 0) {
    out[0] = ws[WS_SUML];
    const float d = tot_trace[0] + ws[WS_REG];
    out[1] = sqrtf(fmaxf(d, 0.0f));
  }
}

// ---------------------------------------------------------------------------
extern "C" void kernel_launch(void* const* d_in, const int* in_sizes, int n_in,
                              void* d_out, int out_size, void* d_ws,
                              size_t ws_size, hipStream_t stream) {
  (void)in_sizes; (void)n_in; (void)out_size;
  const float* mean_pred = (const float*)d_in[0];
  const float* sq_trq    = (const float*)d_in[1];
  const float* tot_trace = (const float*)d_in[2];
  const int*   y         = (const int*)d_in[3];
  float* ws  = (float*)d_ws;
  float* out = (float*)d_out;

  const size_t need = (size_t)(WS_TRQ + (size_t)B_DIM * QP) * sizeof(float);
  const int use_ws_trq = (ws_size >= need) ? 1 : 0;

  hipLaunchKernelGGL(gh_init_kernel, dim3(1), dim3(1), 0, stream, ws);
  if (use_ws_trq) {
    hipLaunchKernelGGL(trq_wmma_kernel, dim3((B_DIM / 16) * (QP / 16)), dim3(32),
                       0, stream, mean_pred, sq_trq, y, ws);
  }
  hipLaunchKernelGGL(gvi_main_kernel, dim3(B_DIM), dim3(256), 0, stream,
                     mean_pred, sq_trq, y, ws, use_ws_trq);
  hipLaunchKernelGGL(finalize_kernel, dim3(1), dim3(1), 0, stream,
                     tot_trace, ws, out);
}